// SSA_Sp_13855564497218
// MI455X (gfx1250) — compile-verified
//
#include <hip/hip_runtime.h>
#include <hip/hip_bf16.h>

// ---- problem constants -----------------------------------------------------
static constexpr int BN  = 8;            // batch
static constexpr int HH  = 96;           // H
static constexpr int WW  = 96;           // W
static constexpr int CC  = 256;          // channels
static constexpr int HC  = HH * CC;      // 24576 = flattened (h,c)/(w,c) row length
static constexpr int SZB = WW * HH * CC; // per-batch q/k/v elements (2359296)
static constexpr size_t SZ = (size_t)BN * SZB; // 18874368 per tensor
static constexpr int NC  = 32;           // split-K chunks for score GEMMs
static constexpr int KCH = HC / NC;      // 768 K per chunk

typedef float v2f __attribute__((ext_vector_type(2)));
typedef float v8f __attribute__((ext_vector_type(8)));

__device__ __forceinline__ v8f wmma_f32(v2f a, v2f b, v8f c) {
    // D = A(16x4 f32) x B(4x16 f32) + C(16x16 f32)
    return __builtin_amdgcn_wmma_f32_16x16x4_f32(false, a, false, b,
                                                 (short)0, c, false, false);
}

// ---------------------------------------------------------------------------
// Kernel 1: QKV GEMM.  qkv[m, o] = sum_c x[m, c] * w[o, c] + bias[o]
// m = (b*W + w)*H + h  (x is (B,W,H,C) row-major), o in [0,768)
// Wave: 16x64 output tile (4 accumulators reuse the A fragment; w_qkv is
// 768 KB -> L2 resident, reused by all 4608 m-tiles).
// grid (4608, 3), block 128 (4 waves).
// ---------------------------------------------------------------------------
__global__ __launch_bounds__(128) void qkv_gemm_k(
    const float* __restrict__ x, const float* __restrict__ w,
    const float* __restrict__ bias,
    float* __restrict__ qW, float* __restrict__ kW, float* __restrict__ vW) {
    const int lane = threadIdx.x & 31;
    const int wv   = threadIdx.x >> 5;
    const int ln   = lane & 15;
    const int lh   = lane >> 4;                  // 0|1 -> K pair offset 0|2
    const int mbase = blockIdx.x * 16;
    const int nb0   = (blockIdx.y * 4 + wv) * 64; // 4 ntiles of 16

    const float* aRow = x + (size_t)(mbase + ln) * CC;
    v8f acc[4] = {};
    for (int kt = 0; kt < CC; kt += 4) {
        v2f a = *(const v2f*)(aRow + kt + 2 * lh);
#pragma unroll
        for (int t = 0; t < 4; ++t) {
            int n  = nb0 + t * 16 + ln;
            v2f bb = *(const v2f*)(w + (size_t)n * CC + kt + 2 * lh);
            acc[t] = wmma_f32(a, bb, acc[t]);
        }
    }
#pragma unroll
    for (int t = 0; t < 4; ++t) {
        int   n   = nb0 + t * 16 + ln;
        float bs  = bias[n];
        int   seg = n >> 8;                       // 0:q 1:k 2:v
        int   col = n & 255;
        float* dst = (seg == 0) ? qW : (seg == 1 ? kW : vW);
#pragma unroll
        for (int r = 0; r < 8; ++r) {
            int m = mbase + r + 8 * lh;
            dst[(size_t)m * CC + col] = acc[t][r] + bs;
        }
    }
}

// ---------------------------------------------------------------------------
// Kernels 2/3: attention-score partials with split-K, K=24576 (dominant WMMA
// count).  2x2 register blocking: each wave owns a 32x32 output tile so each
// A/B fragment feeds two WMMAs -> 1 fragment load per WMMA (vs 2 unblocked).
// MODE 0 (ms): rows are H-major views of the W-major tensors:
//   elem(b,row,j) = (j>>8)*HC + row*C + (j&255)
// MODE 1 (ma): rows contiguous: elem(b,row,j) = row*HC + j
// grid (3, 8, 32), block 96 (3 waves; 9 32x32 tiles per (b,chunk)).
// ---------------------------------------------------------------------------
template <int MODE>
__global__ __launch_bounds__(96) void s_partial_k(
    const float* __restrict__ A, const float* __restrict__ Bm,
    float* __restrict__ Sp) {
    const int lane = threadIdx.x & 31;
    const int wv   = threadIdx.x >> 5;
    const int ln   = lane & 15;
    const int lh   = lane >> 4;
    const int t    = blockIdx.x * 3 + wv;        // 0..8
    const int mt   = t / 3, nt = t % 3;          // 32-wide tile coords
    const int b    = blockIdx.y;
    const int ch   = blockIdx.z;                 // split-K chunk
    const int k0   = ch * KCH;

    const float* Ab = A  + (size_t)b * SZB;
    const float* Bb = Bm + (size_t)b * SZB;
    const int m0 = mt * 32 + ln;                 // A rows m0, m0+16
    const int n0 = nt * 32 + ln;                 // B cols n0, n0+16

    v8f acc[2][2] = {};
    for (int kt = k0; kt < k0 + KCH; kt += 4) {
        int j = kt + 2 * lh;
        v2f a0, a1, b0, b1;
        if (MODE == 0) {
            int w_ = j >> 8, c_ = j & 255;
            const float* pa = Ab + (size_t)w_ * HC + c_;
            const float* pb = Bb + (size_t)w_ * HC + c_;
            a0 = *(const v2f*)(pa + (size_t)m0 * CC);
            a1 = *(const v2f*)(pa + (size_t)(m0 + 16) * CC);
            b0 = *(const v2f*)(pb + (size_t)n0 * CC);
            b1 = *(const v2f*)(pb + (size_t)(n0 + 16) * CC);
        } else {
            a0 = *(const v2f*)(Ab + (size_t)m0 * HC + j);
            a1 = *(const v2f*)(Ab + (size_t)(m0 + 16) * HC + j);
            b0 = *(const v2f*)(Bb + (size_t)n0 * HC + j);
            b1 = *(const v2f*)(Bb + (size_t)(n0 + 16) * HC + j);
        }
        acc[0][0] = wmma_f32(a0, b0, acc[0][0]);
        acc[0][1] = wmma_f32(a0, b1, acc[0][1]);
        acc[1][0] = wmma_f32(a1, b0, acc[1][0]);
        acc[1][1] = wmma_f32(a1, b1, acc[1][1]);
    }
    float* ob = Sp + ((size_t)ch * BN + b) * 96 * 96;
#pragma unroll
    for (int um = 0; um < 2; ++um)
#pragma unroll
        for (int un = 0; un < 2; ++un) {
            float* o = ob + (size_t)(mt * 32 + um * 16) * 96 + nt * 32 + un * 16;
#pragma unroll
            for (int r = 0; r < 8; ++r)
                o[(r + 8 * lh) * 96 + ln] = acc[um][un][r];
        }
}

// ---------------------------------------------------------------------------
// Kernel 4: deterministic split-K reduction + row softmax (96-wide rows).
// One wave per row; lane owns cols {lane, lane+32, lane+64}.
// grid 768 blocks of 32 threads.
// ---------------------------------------------------------------------------
__global__ __launch_bounds__(32) void softmax_rows_k(
    const float* __restrict__ Sp, float* __restrict__ S) {
    const int lane = threadIdx.x;
    const int row  = blockIdx.x;                 // b*96 + r, 0..767
    float v0 = 0.f, v1 = 0.f, v2 = 0.f;
    for (int c = 0; c < NC; ++c) {
        const float* p = Sp + ((size_t)c * (BN * 96) + row) * 96;
        v0 += p[lane]; v1 += p[lane + 32]; v2 += p[lane + 64];
    }
    float m = fmaxf(fmaxf(v0, v1), v2);
    for (int off = 16; off; off >>= 1) m = fmaxf(m, __shfl_xor(m, off, 32));
    float e0 = __expf(v0 - m), e1 = __expf(v1 - m), e2 = __expf(v2 - m);
    float s = e0 + e1 + e2;
    for (int off = 16; off; off >>= 1) s += __shfl_xor(s, off, 32);
    float inv = 1.0f / s;
    float* o = S + (size_t)row * 96;
    o[lane] = e0 * inv; o[lane + 32] = e1 * inv; o[lane + 64] = e2 * inv;
}

// ---------------------------------------------------------------------------
// Kernels 5/6: apply attention.  D[b,row,j] = sum_k S[b,row,k] * V(b,k,j), K=96.
// Two M-tiles per wave so each (strided-b32) B fragment feeds two WMMAs.
// MODE 0 (ms): V in H-major view; write '=' to out[((b*W + j>>8)*H + row)*C + j&255]
// MODE 1 (ma): V rows contiguous;  '+=' to out[(b*W + row)*HC + j]
// grid (384, 3, 8), block 128 (4 waves, each a 32x16 tile).
// ---------------------------------------------------------------------------
template <int MODE>
__global__ __launch_bounds__(128) void attn_out_k(
    const float* __restrict__ S, const float* __restrict__ V,
    float* __restrict__ out) {
    const int lane = threadIdx.x & 31;
    const int wv   = threadIdx.x >> 5;
    const int ln   = lane & 15;
    const int lh   = lane >> 4;
    const int nt   = blockIdx.x * 4 + wv;        // 0..1535
    const int mt   = blockIdx.y;                 // 0..2 (pair of 16-row tiles)
    const int b    = blockIdx.z;

    const float* Sb = S + (size_t)b * 96 * 96;
    const float* Vb = V + (size_t)b * SZB;
    const int m0 = mt * 32 + ln;                 // rows m0, m0+16
    const int n  = nt * 16 + ln;                 // j

    v8f acc[2] = {};
    for (int kt = 0; kt < 96; kt += 4) {
        int kb = kt + 2 * lh;
        v2f a0 = *(const v2f*)(Sb + (size_t)m0 * 96 + kb);
        v2f a1 = *(const v2f*)(Sb + (size_t)(m0 + 16) * 96 + kb);
        v2f bb;
        if (MODE == 0) {
            int w_ = n >> 8, c_ = n & 255;
            const float* p = Vb + (size_t)w_ * HC + (size_t)kb * CC + c_;
            bb = v2f{p[0], p[CC]};
        } else {
            const float* p = Vb + (size_t)kb * HC + n;
            bb = v2f{p[0], p[HC]};
        }
        acc[0] = wmma_f32(a0, bb, acc[0]);
        acc[1] = wmma_f32(a1, bb, acc[1]);
    }
    if (MODE == 0) {
        int w_ = n >> 8, c_ = n & 255;
#pragma unroll
        for (int u = 0; u < 2; ++u)
#pragma unroll
            for (int r = 0; r < 8; ++r) {
                int h = mt * 32 + u * 16 + r + 8 * lh;
                out[(((size_t)b * WW + w_) * HH + h) * CC + c_] = acc[u][r];
            }
    } else {
#pragma unroll
        for (int u = 0; u < 2; ++u)
#pragma unroll
            for (int r = 0; r < 8; ++r) {
                int w1 = mt * 32 + u * 16 + r + 8 * lh;
                float* p = out + ((size_t)b * WW + w1) * HC + n;
                *p += acc[u][r];
            }
    }
}

// ---------------------------------------------------------------------------
extern "C" void kernel_launch(void* const* d_in, const int* in_sizes, int n_in,
                              void* d_out, int out_size, void* d_ws, size_t ws_size,
                              hipStream_t stream) {
    const float* x    = (const float*)d_in[0];   // (B, W, H, C)
    const float* w    = (const float*)d_in[1];   // (768, 256)
    const float* bias = (const float*)d_in[2];   // (768,)
    float* out = (float*)d_out;                  // (B, W, H, C)

    float* ws   = (float*)d_ws;
    float* qW   = ws;                            // (b,w,h,c)
    float* kW   = qW + SZ;
    float* vW   = kW + SZ;
    float* SpMs = vW + SZ;                       // NC x B x 96 x 96 partials
    float* SpMa = SpMs + (size_t)NC * BN * 96 * 96;
    float* Sms  = SpMa + (size_t)NC * BN * 96 * 96;  // B x 96 x 96 softmaxed
    float* Sma  = Sms + (size_t)BN * 96 * 96;

    // 1) QKV GEMM (M=73728, N=768, K=256)
    qkv_gemm_k<<<dim3((BN * WW * HH) / 16, 3, 1), 128, 0, stream>>>(
        x, w, bias, qW, kW, vW);
    // 2) score partials: ms uses q rows x k rows (H-major view)
    s_partial_k<0><<<dim3(3, BN, NC), 96, 0, stream>>>(qW, kW, SpMs);
    //    ma uses k rows x q rows (W-major, contiguous)
    s_partial_k<1><<<dim3(3, BN, NC), 96, 0, stream>>>(kW, qW, SpMa);
    // 3) deterministic reduce + softmax
    softmax_rows_k<<<dim3(BN * 96), 32, 0, stream>>>(SpMs, Sms);
    softmax_rows_k<<<dim3(BN * 96), 32, 0, stream>>>(SpMa, Sma);
    // 4) apply: ms writes '=', ma accumulates '+=' (stream-ordered, deterministic)
    attn_out_k<0><<<dim3(384, 3, BN), 128, 0, stream>>>(Sms, vW, out);
    attn_out_k<1><<<dim3(384, 3, BN), 128, 0, stream>>>(Sma, vW, out);
}